// WeightedChamferLoss_12506944766656
// MI455X (gfx1250) — compile-verified
//
#include <hip/hip_runtime.h>
#include <hip/hip_bf16.h>
#include <math.h>

typedef __attribute__((ext_vector_type(2))) float v2f;
typedef __attribute__((ext_vector_type(8))) float v8f;

// ---- order-preserving float<->uint mapping (valid for ALL floats incl. negatives) ----
__device__ __forceinline__ unsigned f2ord(float f) {
    unsigned u = __float_as_uint(f);
    return (u & 0x80000000u) ? ~u : (u | 0x80000000u);
}
__device__ __forceinline__ float ord2f(unsigned u) {
    u = (u & 0x80000000u) ? (u & 0x7FFFFFFFu) : ~u;
    return __uint_as_float(u);
}
#define ORD_POS_INF 0xFF800000u   // f2ord(+inf)

// ================= init colmin buffers to encoded +inf =================
__global__ void init_min_kernel(unsigned* __restrict__ p, int n) {
    int i = blockIdx.x * blockDim.x + threadIdx.x;
    if (i < n) p[i] = ORD_POS_INF;
}

// ================= fused pairwise sqdist + row/col min =================
// Apts: [B, Nrows, 3] (rows), Bpts: [B, Mcols, 3] (cols)
// rowmin: [B, Nrows] plain float (wave owns full rows -> exact, plain store)
// colmin: [B, Mcols] encoded uint, combined across row-blocks via atomicMin
// One wave = 32 rows (two 16-row A fragments) x ALL columns.
// Per 16-col tile: 1 B-point load (software-pipelined) -> 2 WMMAs -> 1 atomic/col.
// Requires Nrows % (32*wavesPerBlock) == 0, Mcols % 16 == 0.
__global__ void pairwise_min_kernel(const float* __restrict__ Apts,
                                    const float* __restrict__ Bpts,
                                    float* __restrict__ rowmin,
                                    unsigned* __restrict__ colmin,
                                    int Nrows, int Mcols) {
    const int b    = blockIdx.y;
    const int lane = threadIdx.x & 31;
    const int wave = threadIdx.x >> 5;
    const int wavesPerBlock = blockDim.x >> 5;
    const int n0 = (blockIdx.x * wavesPerBlock + wave) * 32;
    if (n0 >= Nrows) return;  // wave-uniform; EXEC stays all-ones inside

    const float* A  = Apts + (size_t)b * Nrows * 3;
    const float* Bp = Bpts + (size_t)b * Mcols * 3;
    const bool hi = (lane >= 16);
    const int  l15 = lane & 15;

    // ---- two A fragments: rows n0+(lane&15) and n0+16+(lane&15) ----
    // A row K-vector = (x, y, z, ||a||^2); lanes 0-15 carry K0,K1; lanes 16-31 K2,K3.
    v2f afrag0, afrag1;
    {
        const int ar = n0 + l15;
        const float ax = A[ar * 3 + 0], ay = A[ar * 3 + 1], az = A[ar * 3 + 2];
        const float an = ax * ax + ay * ay + az * az;
        afrag0.x = hi ? az : ax;
        afrag0.y = hi ? an : ay;
    }
    {
        const int ar = n0 + 16 + l15;
        const float ax = A[ar * 3 + 0], ay = A[ar * 3 + 1], az = A[ar * 3 + 2];
        const float an = ax * ax + ay * ay + az * az;
        afrag1.x = hi ? az : ax;
        afrag1.y = hi ? an : ay;
    }

    const float INF = __builtin_inff();
    float rmin0[8], rmin1[8];
#pragma unroll
    for (int r = 0; r < 8; ++r) { rmin0[r] = INF; rmin1[r] = INF; }

    // ---- software-pipelined sweep over 16-wide column tiles ----
    float bx = Bp[l15 * 3 + 0];
    float by = Bp[l15 * 3 + 1];
    float bz = Bp[l15 * 3 + 2];

    for (int m0 = 0; m0 < Mcols; m0 += 16) {
        const int mc = m0 + l15;

        // build B fragment from current tile's point: K = (-2bx, -2by, -2bz, 1)
        const float bn = bx * bx + by * by + bz * bz;   // per-column constant
        v2f bfrag;
        bfrag.x = hi ? (-2.0f * bz) : (-2.0f * bx);
        bfrag.y = hi ? 1.0f : (-2.0f * by);

        // prefetch next tile's point (wraps to tile 0 on last iter; harmless)
        const int mnext = (m0 + 16 < Mcols) ? (m0 + 16 + l15) : l15;
        const float nbx = Bp[mnext * 3 + 0];
        const float nby = Bp[mnext * 3 + 1];
        const float nbz = Bp[mnext * 3 + 2];

        // D[n][m] = ||a_n||^2 - 2 a_n.b_m  (full f32, K=4); two row-blocks share B
        v8f c0 = {0.f, 0.f, 0.f, 0.f, 0.f, 0.f, 0.f, 0.f};
        v8f c1 = {0.f, 0.f, 0.f, 0.f, 0.f, 0.f, 0.f, 0.f};
        c0 = __builtin_amdgcn_wmma_f32_16x16x4_f32(false, afrag0, false, bfrag,
                                                   (short)0, c0, false, false);
        c1 = __builtin_amdgcn_wmma_f32_16x16x4_f32(false, afrag1, false, bfrag,
                                                   (short)0, c1, false, false);

        float cm = INF;   // min over this lane's 16 rows (both blocks) of its column
#pragma unroll
        for (int r = 0; r < 8; ++r) {
            const float d0 = c0[r] + bn;            // full squared distance
            const float d1 = c1[r] + bn;
            rmin0[r] = fminf(rmin0[r], d0);
            rmin1[r] = fminf(rmin1[r], d1);
            cm = fminf(cm, fminf(d0, d1));
        }
        // combine half-waves (rows 0-7 vs 8-15 within each block share a column)
        cm = fminf(cm, __shfl_xor(cm, 16, 32));
        if (lane < 16) {
            atomicMin(&colmin[(size_t)b * Mcols + mc], f2ord(cm));
        }

        bx = nbx; by = nby; bz = nbz;
    }

    // ---- finish row mins: reduce across the 16 lanes of each half-wave ----
#pragma unroll
    for (int r = 0; r < 8; ++r) {
        float v0 = rmin0[r];
        v0 = fminf(v0, __shfl_xor(v0, 1, 32));
        v0 = fminf(v0, __shfl_xor(v0, 2, 32));
        v0 = fminf(v0, __shfl_xor(v0, 4, 32));
        v0 = fminf(v0, __shfl_xor(v0, 8, 32));
        rmin0[r] = v0;
        float v1 = rmin1[r];
        v1 = fminf(v1, __shfl_xor(v1, 1, 32));
        v1 = fminf(v1, __shfl_xor(v1, 2, 32));
        v1 = fminf(v1, __shfl_xor(v1, 4, 32));
        v1 = fminf(v1, __shfl_xor(v1, 8, 32));
        rmin1[r] = v1;
    }
    if (l15 == 0) {  // lane 0 -> rows +0..7, lane 16 -> rows +8..15 (per block)
        const int rowbase = n0 + (hi ? 8 : 0);
        float* dst0 = rowmin + (size_t)b * Nrows + rowbase;
        float* dst1 = dst0 + 16;
#pragma unroll
        for (int r = 0; r < 8; ++r) { dst0[r] = rmin0[r]; dst1[r] = rmin1[r]; }
    }
}

// ================= final fused reduction (single workgroup) =================
__global__ void finalize_kernel(const float* __restrict__ rowmin1,       // [B,N]
                                const unsigned* __restrict__ colmin1,    // [B,M]
                                const float* __restrict__ rowmin2,       // [B,P]
                                const unsigned* __restrict__ colmin2,    // [B,N]
                                const float* __restrict__ weight_map,    // [B,C,M]
                                float* __restrict__ out,
                                int B, int N, int M, int P, int C) {
    const int tid = threadIdx.x;
    const int nt  = blockDim.x;
    const float inv_bn = 1.0f / (float)(B * N);
    const float inv_bm = 1.0f / (float)(B * M);
    const float inv_bp = 1.0f / (float)(B * P);

    float acc = 0.0f;
    for (int i = tid; i < B * N; i += nt) acc += rowmin1[i] * inv_bn;
    for (int i = tid; i < B * M; i += nt) {
        const int b = i / M, m = i - b * M;
        float s = 0.0f;
        for (int c = 0; c < C; ++c) s += weight_map[((size_t)b * C + c) * M + m];
        const float w = 1.0f / (1.0f + expf(-s)) + 1.0f;   // sigmoid + 1
        acc += ord2f(colmin1[i]) * w * inv_bm;
    }
    for (int i = tid; i < B * P; i += nt) acc += rowmin2[i] * inv_bp;
    for (int i = tid; i < B * N; i += nt) acc += ord2f(colmin2[i]) * inv_bn;

    __shared__ float sdata[256];
    sdata[tid] = acc;
    __syncthreads();
    for (int s = nt >> 1; s > 0; s >>= 1) {
        if (tid < s) sdata[tid] += sdata[tid + s];
        __syncthreads();
    }
    if (tid == 0) out[0] = sdata[0];
}

// ================= host launcher =================
extern "C" void kernel_launch(void* const* d_in, const int* in_sizes, int n_in,
                              void* d_out, int out_size, void* d_ws, size_t ws_size,
                              hipStream_t stream) {
    // Reference shapes: B=8, N=2048, M=2048, P=4096, D=3, C=3
    const int B = 8, N = 2048, M = 2048, P = 4096, C = 3;

    const float* fps       = (const float*)d_in[0];   // [B,M,3]
    const float* structure = (const float*)d_in[1];   // [B,N,3]
    const float* wmap      = (const float*)d_in[2];   // [B,C,M]
    const float* origin    = (const float*)d_in[3];   // [B,P,3]
    float* out = (float*)d_out;

    // workspace layout (4B elems): rowmin1 | colmin1 | rowmin2 | colmin2
    float*    rowmin1 = (float*)d_ws;                       // B*N
    unsigned* colmin1 = (unsigned*)(rowmin1 + (size_t)B*N); // B*M
    float*    rowmin2 = (float*)(colmin1 + (size_t)B*M);    // B*P
    unsigned* colmin2 = (unsigned*)(rowmin2 + (size_t)B*P); // B*N

    // 1) init colmin buffers to encoded +inf
    {
        const int n1 = B * M;
        init_min_kernel<<<(n1 + 255) / 256, 256, 0, stream>>>(colmin1, n1);
        const int n2 = B * N;
        init_min_kernel<<<(n2 + 255) / 256, 256, 0, stream>>>(colmin2, n2);
    }

    // 2) end-distance: structure (rows, N) vs fps (cols, M)
    {
        dim3 block(128);                 // 4 waves; each wave = 32 rows
        dim3 grid(N / 128, B);           // (rows/32)/4 blocks per batch
        pairwise_min_kernel<<<grid, block, 0, stream>>>(structure, fps,
                                                        rowmin1, colmin1, N, M);
    }

    // 3) chamfer: origin (rows, P) vs structure (cols, N)
    {
        dim3 block(128);
        dim3 grid(P / 128, B);
        pairwise_min_kernel<<<grid, block, 0, stream>>>(origin, structure,
                                                        rowmin2, colmin2, P, N);
    }

    // 4) fused sigmoid-weight + four means + scalar output
    finalize_kernel<<<1, 256, 0, stream>>>(rowmin1, colmin1, rowmin2, colmin2,
                                           wmap, out, B, N, M, P, C);
}